// VectorQuantize_44796508897704
// MI455X (gfx1250) — compile-verified
//
#include <hip/hip_runtime.h>
#include <hip/hip_bf16.h>
#include <math.h>

typedef float v2f __attribute__((ext_vector_type(2)));
typedef float v8f __attribute__((ext_vector_type(8)));
typedef unsigned int v4u __attribute__((ext_vector_type(4)));
typedef int v8i_t __attribute__((ext_vector_type(8)));
typedef int v4i_t __attribute__((ext_vector_type(4)));

// ---------------------------------------------------------------------------
// Tensor Data Mover: 2D tile (global, row-major, strided) -> LDS (contiguous,
// optionally padded every pad_interval DWORDs by pad_amount DWORDs).
// Descriptor packing per CDNA5 ISA 8.3/8.4 (D# group0 + group1, 2D tensor,
// groups 2/3 zero). data_size = 4 bytes.
// 6-arg builtin form: (uint32x4 g0, int32x8 g1, int32x4, int32x4, int32x8, i32 cpol)
//   tile_d0 : contiguous extent (elements) along dim0
//   tile_d1 : number of lines
//   stride0 : elements between consecutive lines
// ---------------------------------------------------------------------------
__device__ __forceinline__ void tdm_load_2d(
    unsigned int lds_byte_addr, const void* gptr,
    unsigned int tile_d0, unsigned int tile_d1, unsigned long long stride0,
    unsigned int pad_interval_code, unsigned int pad_amount_code,
    unsigned int pad_enable)
{
    unsigned long long ga = (unsigned long long)(uintptr_t)gptr;

    v4u g0;
    g0.x = 1u;                                     // count=1, is_restore=0
    g0.y = lds_byte_addr;                          // lds_addr [63:32]
    g0.z = (unsigned int)(ga & 0xFFFFFFFFu);       // global_addr[31:0]
    g0.w = (unsigned int)((ga >> 32) & 0x01FFFFFFu)// global_addr[56:32]
         | 0x80000000u;                            // type=2 ("image") [127:126]

    unsigned int w0 = (2u << 16)                   // data_size = 4 bytes
                    | (pad_enable ? (1u << 20) : 0u)
                    | (pad_interval_code << 22)
                    | (pad_amount_code << 25);
    unsigned int td0 = tile_d0, td1 = tile_d1;     // tensor dims == tile dims
    v8i_t g1;
    g1[0] = (int)w0;
    g1[1] = (int)((td0 & 0xFFFFu) << 16);          // tensor_dim0[15:0] @ [63:48]
    g1[2] = (int)(((td0 >> 16) & 0xFFFFu) | ((td1 & 0xFFFFu) << 16));
    g1[3] = (int)(((td1 >> 16) & 0xFFFFu) | ((tile_d0 & 0xFFFFu) << 16));
    g1[4] = (int)(tile_d1 & 0xFFFFu);              // tile_dim1; tile_dim2=0
    g1[5] = (int)(unsigned int)(stride0 & 0xFFFFFFFFu);
    g1[6] = (int)(unsigned int)((stride0 >> 32) & 0xFFFFu);
    g1[7] = 0;

    v4i_t z4 = {0, 0, 0, 0};
    v8i_t z8 = {0, 0, 0, 0, 0, 0, 0, 0};
    __builtin_amdgcn_tensor_load_to_lds(g0, g1, z4, z4, z8, 0);
}

// ---------------------------------------------------------------------------
// Weight-norm / L2-norm row scaling: out[r,:] = scale(r) * v[r,:]
//   use_eps==0 : scale = g[r] / ||v_r||        (PyTorch weight_norm)
//   use_eps==1 : scale = 1 / max(||v_r||, eps) (F.normalize)
// One wave (32 lanes) per row.
// ---------------------------------------------------------------------------
__global__ __launch_bounds__(256) void rownorm_scale(
    const float* __restrict__ v, const float* __restrict__ g,
    float* __restrict__ out, int rows, int cols, int use_eps)
{
    const int wavesPerBlock = blockDim.x >> 5;
    const int row  = blockIdx.x * wavesPerBlock + (threadIdx.x >> 5);
    const int lane = threadIdx.x & 31;
    if (row >= rows) return;

    const float* src = v + (size_t)row * cols;
    float s = 0.f;
    for (int c = lane; c < cols; c += 32) { float x = src[c]; s += x * x; }
    #pragma unroll
    for (int m = 16; m >= 1; m >>= 1) s += __shfl_xor(s, m, 32);

    float norm = sqrtf(s);
    float scale;
    if (use_eps) { norm = fmaxf(norm, 1e-12f); scale = 1.0f / norm; }
    else         { scale = g[row] / norm; }

    float* dst = out + (size_t)row * cols;
    for (int c = lane; c < cols; c += 32) dst[c] = src[c] * scale;
}

// ---------------------------------------------------------------------------
// f32 WMMA GEMM with bias:  C[b] = A @ B[b] + bias  (row-major everywhere)
// Block = 128 threads (4 waves). Macro-tile 64(M) x 64(N), BK = 16.
// Double-buffered TDM pipeline: while waves compute on buffer p, wave 0 has
// already launched the DMA for tile i+1 into buffer p^1. s_wait_tensorcnt(2)
// exploits TDM in-order completion: <=2 outstanding means tile i's pair is
// done while tile i+1's pair stays in flight.
//   Alds : 64 lines x 16 k, TDM-padded to stride 18 (aligned b64 frag reads)
//   Blds : 16 lines x 64 n, contiguous (k-major)
// V_WMMA_F32_16X16X4_F32 fragment layout (ISA 7.12.2):
//   A 16x4 : lane L -> row L%16 ; vgpr j -> K = (L<16 ? j : j+2)
//   B 4x16 : lane L -> col L%16 ; vgpr j -> K = (L<16 ? j : j+2)
//   C 16x16: vgpr r -> row r (lanes 0-15) / r+8 (lanes 16-31), col = L%16
// ---------------------------------------------------------------------------
__global__ __launch_bounds__(128) void wmma_gemm_bias(
    const float* __restrict__ A, const float* __restrict__ Bmat,
    const float* __restrict__ bias, float* __restrict__ C,
    int M, int N, int K)
{
    const int b  = blockIdx.z;
    const float* Bb = Bmat + (size_t)b * K * N;
    float*       Cb = C    + (size_t)b * M * N;
    const int m0 = blockIdx.y * 64;
    const int n0 = blockIdx.x * 64;

    __shared__ float Alds[2][64 * 18];   // [m][k], padded rows (16 data + 2 pad)
    __shared__ float Blds[2][16 * 64];   // [k][n]

    const int tid  = threadIdx.x;
    const int wave = tid >> 5;
    const int lane = tid & 31;
    const int half = lane >> 4;
    const int l16  = lane & 15;
    const int mloc = wave * 16 + l16;

    const unsigned int aAddr0 = (unsigned int)(uintptr_t)(&Alds[0][0]);
    const unsigned int aAddr1 = (unsigned int)(uintptr_t)(&Alds[1][0]);
    const unsigned int bAddr0 = (unsigned int)(uintptr_t)(&Blds[0][0]);
    const unsigned int bAddr1 = (unsigned int)(uintptr_t)(&Blds[1][0]);

    const int ntiles = K >> 4;

    // prologue: DMA tile 0 into buffer 0
    if (wave == 0) {
        tdm_load_2d(aAddr0, A + (size_t)m0 * K, 16, 64,
                    (unsigned long long)K, 3, 1, 1);
        tdm_load_2d(bAddr0, Bb + n0, 64, 16,
                    (unsigned long long)N, 0, 0, 0);
    }

    v8f acc[4] = {};

    for (int i = 0; i < ntiles; ++i) {
        const int p = i & 1;
        __syncthreads();                    // everyone done reading buf p^1
        if (i + 1 < ntiles) {
            if (wave == 0) {
                const int k1 = (i + 1) << 4;
                const unsigned int aA = p ? aAddr0 : aAddr1;
                const unsigned int bA = p ? bAddr0 : bAddr1;
                tdm_load_2d(aA, A + (size_t)m0 * K + k1, 16, 64,
                            (unsigned long long)K, 3, 1, 1);
                tdm_load_2d(bA, Bb + (size_t)k1 * N + n0, 64, 16,
                            (unsigned long long)N, 0, 0, 0);
            }
            __builtin_amdgcn_s_wait_tensorcnt(2);  // tile i's pair complete
        } else {
            __builtin_amdgcn_s_wait_tensorcnt(0);
        }
        __syncthreads();                    // buf p visible to all waves

        const float* Ap = &Alds[p][0];
        const float* Bp = &Blds[p][0];
        #pragma unroll
        for (int kk = 0; kk < 16; kk += 4) {
            const int ka = kk + (half ? 2 : 0);
            v2f a;
            a.x = Ap[mloc * 18 + ka];
            a.y = Ap[mloc * 18 + ka + 1];
            #pragma unroll
            for (int nt = 0; nt < 4; ++nt) {
                const int n = nt * 16 + l16;
                v2f bv;
                bv.x = Bp[ka * 64 + n];
                bv.y = Bp[(ka + 1) * 64 + n];
                acc[nt] = __builtin_amdgcn_wmma_f32_16x16x4_f32(
                    false, a, false, bv, (short)0, acc[nt], false, false);
            }
        }
    }

    const int mbase = m0 + wave * 16 + (half ? 8 : 0);
    #pragma unroll
    for (int r = 0; r < 8; ++r) {
        const int m = mbase + r;
        const float bvl = bias[m];
        #pragma unroll
        for (int nt = 0; nt < 4; ++nt)
            Cb[(size_t)m * N + n0 + nt * 16 + l16] = acc[nt][r] + bvl;
    }
}

// ---------------------------------------------------------------------------
// Fused cosine-similarity GEMM + row argmax.
//   scores[t, j] = sum_c z_i[b, c, t] * cbn[j, c]   (cbn = L2-normalized codebook)
//   idx[t] = argmax_j scores[t, j]   (first-max tie-break, like jnp.argmax)
// SINGLE-WAVE blocks (32 threads, 16 time-rows): no barriers needed at all.
// enc tile staged once by TDM (256 lines x 16 t-elems, stride T = strided
// transpose gather). Codebook tiles (16 codes x 256, TDM-padded to stride 260)
// are double-buffered: prefetch tile i+1, s_wait_tensorcnt(1), compute tile i.
// ---------------------------------------------------------------------------
__global__ __launch_bounds__(32) void argmax_codes(
    const float* __restrict__ z_i,   // (B, 256, T)
    const float* __restrict__ cbn,   // (4096, 256) normalized
    int*   __restrict__ idx_out,     // (B*T)
    float* __restrict__ idx_out_f,   // indices region of d_out (as float)
    int T)
{
    constexpr int D = 256, KCB = 4096, ROWS = 16, CBPAD = 260;
    const int r0 = blockIdx.x * ROWS;          // flat row (b*T + t); 16 | T
    const int bIdx = r0 / T;
    const int t0   = r0 % T;
    const float* zb = z_i + (size_t)bIdx * D * T;

    __shared__ float enc[D * ROWS];            // [c][m], lines of 16 (16 KB)
    __shared__ float cb[2][16 * CBPAD];        // [n][c], padded (2 x 16.6 KB)

    const int lane = threadIdx.x;              // one full wave
    const int half = lane >> 4;
    const int l16  = lane & 15;

    const unsigned int encAddr = (unsigned int)(uintptr_t)(&enc[0]);
    const unsigned int cbAddr0 = (unsigned int)(uintptr_t)(&cb[0][0]);
    const unsigned int cbAddr1 = (unsigned int)(uintptr_t)(&cb[1][0]);

    // prologue: enc tile + first codebook tile
    tdm_load_2d(encAddr, zb + t0, ROWS, D, (unsigned long long)T, 0, 0, 0);
    tdm_load_2d(cbAddr0, cbn, D, 16, (unsigned long long)D, 7, 3, 1);

    float best[8];
    int   bidx[8];
    #pragma unroll
    for (int r = 0; r < 8; ++r) { best[r] = -3.4e38f; bidx[r] = 0; }

    constexpr int NT = KCB / 16;               // 256 codebook tiles
    for (int i = 0; i < NT; ++i) {
        const int p = i & 1;
        if (i + 1 < NT) {
            // prefetch next tile into the idle buffer; previous compute's LDS
            // reads have already completed (their values fed this wave's WMMAs)
            tdm_load_2d(p ? cbAddr0 : cbAddr1, cbn + (size_t)(i + 1) * 16 * D,
                        D, 16, (unsigned long long)D, 7, 3, 1);
            __builtin_amdgcn_s_wait_tensorcnt(1);  // tile i (and enc) done
        } else {
            __builtin_amdgcn_s_wait_tensorcnt(0);
        }

        const float* cbp = &cb[p][0];
        v8f acc = {};
        #pragma unroll 4
        for (int k = 0; k < D; k += 4) {
            const int ke = k + (half ? 2 : 0);
            v2f a;
            a.x = enc[ke * ROWS + l16];
            a.y = enc[(ke + 1) * ROWS + l16];
            v2f bv;
            bv.x = cbp[l16 * CBPAD + ke];
            bv.y = cbp[l16 * CBPAD + ke + 1];
            acc = __builtin_amdgcn_wmma_f32_16x16x4_f32(
                false, a, false, bv, (short)0, acc, false, false);
        }

        const int code = i * 16 + l16;
        #pragma unroll
        for (int r = 0; r < 8; ++r) {
            if (acc[r] > best[r]) { best[r] = acc[r]; bidx[r] = code; }
        }
    }

    // reduce over the 16 lanes of each half-wave (first-max wins on ties)
    #pragma unroll
    for (int s = 1; s < 16; s <<= 1) {
        #pragma unroll
        for (int r = 0; r < 8; ++r) {
            float ob = __shfl_xor(best[r], s, 32);
            int   oi = __shfl_xor(bidx[r], s, 32);
            if (ob > best[r] || (ob == best[r] && oi < bidx[r])) {
                best[r] = ob; bidx[r] = oi;
            }
        }
    }

    if (l16 == 0) {
        #pragma unroll
        for (int r = 0; r < 8; ++r) {
            const int m = (half ? 8 : 0) + r;   // rows 0-7 / 8-15
            const int flat = r0 + m;
            idx_out[flat]   = bidx[r];
            idx_out_f[flat] = (float)bidx[r];
        }
    }
}

// ---------------------------------------------------------------------------
// z_q[b, c, t] = codebook[idx[b*T + t], c]; writes coalesced over t.
// ---------------------------------------------------------------------------
__global__ __launch_bounds__(256) void gather_zq(
    const float* __restrict__ codebook, const int* __restrict__ idx,
    float* __restrict__ zq, int T, int D)
{
    const int r0 = blockIdx.x * 64;
    const int b  = r0 / T;
    const int t0 = r0 % T;
    const int tid = threadIdx.x;
    const int tt  = tid & 63;
    const int c0  = tid >> 6;                  // 0..3

    const int code = idx[r0 + tt];
    const float* src = codebook + (size_t)code * D;
    float* dst = zq + (size_t)b * D * T + (t0 + tt);
    for (int c = c0; c < D; c += 4)
        dst[(size_t)c * T] = src[c];
}

// ---------------------------------------------------------------------------
extern "C" void kernel_launch(void* const* d_in, const int* in_sizes, int n_in,
                              void* d_out, int out_size, void* d_ws, size_t ws_size,
                              hipStream_t stream) {
    constexpr int B = 16, DIN = 1024, T = 2048, DCB = 256, KCB = 4096;

    const float* z        = (const float*)d_in[0];
    const float* in_v     = (const float*)d_in[1];
    const float* in_g     = (const float*)d_in[2];
    const float* in_b     = (const float*)d_in[3];
    const float* out_v    = (const float*)d_in[4];
    const float* out_g    = (const float*)d_in[5];
    const float* out_b    = (const float*)d_in[6];
    const float* codebook = (const float*)d_in[7];

    // workspace layout (floats)
    float* ws    = (float*)d_ws;
    float* W_in  = ws;                                  // 256*1024
    float* W_out = W_in + (size_t)DCB * DIN;            // 1024*256
    float* cbn   = W_out + (size_t)DIN * DCB;           // 4096*256
    int*   idx   = (int*)(cbn + (size_t)KCB * DCB);     // 16*2048

    // output layout (floats): z_i | z_q | z_o | indices
    float* z_i   = (float*)d_out;
    float* z_q   = z_i + (size_t)B * DCB * T;           // +8388608
    float* z_o   = z_q + (size_t)B * DCB * T;           // +16777216
    float* ind_f = z_o + (size_t)B * DIN * T;           // +50331648

    // 1) normalized weights + codebook
    rownorm_scale<<<DCB / 8, 256, 0, stream>>>(in_v,  in_g,  W_in,  DCB, DIN, 0);
    rownorm_scale<<<DIN / 8, 256, 0, stream>>>(out_v, out_g, W_out, DIN, DCB, 0);
    rownorm_scale<<<KCB / 8, 256, 0, stream>>>(codebook, nullptr, cbn, KCB, DCB, 1);

    // 2) z_i = W_in @ z + in_b   (per batch: 256 x 2048 x 1024)
    {
        dim3 grid(T / 64, DCB / 64, B);
        wmma_gemm_bias<<<grid, 128, 0, stream>>>(W_in, z, in_b, z_i, DCB, T, DIN);
    }

    // 3) fused similarity GEMM + argmax (32768 x 4096 x 256)
    argmax_codes<<<(B * T) / 16, 32, 0, stream>>>(z_i, cbn, idx, ind_f, T);

    // 4) z_q = codebook[idx]
    gather_zq<<<(B * T) / 64, 256, 0, stream>>>(codebook, idx, z_q, T, DCB);

    // 5) z_o = W_out @ z_q + out_b   (per batch: 1024 x 2048 x 256)
    {
        dim3 grid(T / 64, DIN / 64, B);
        wmma_gemm_bias<<<grid, 128, 0, stream>>>(W_out, z_q, out_b, z_o, DIN, T, DCB);
    }
}